// LabelSmoothingLoss_7971459301814
// MI455X (gfx1250) — compile-verified
//
#include <hip/hip_runtime.h>
#include <hip/hip_bf16.h>
#include <math.h>

// ---------------- problem constants ----------------
#define N_TOKENS   8192
#define VOCAB      32000
#define VOCAB4     (VOCAB / 4)        // 8000 float4 per row
#define SMOOTHING  0.1f
#define CONFIDENCE 0.9f
#define EPS_SMOOTH (SMOOTHING / (float)(VOCAB - 2))
#define BLOCK1     320                // 10 wave32s: 8000/320 = 25 float4/thread exactly
#define NWAVES1    (BLOCK1 / 32)
#define BLOCK2     256

typedef float v8f __attribute__((ext_vector_type(8)));
typedef float v2f __attribute__((ext_vector_type(2)));

// ---------------------------------------------------------------------------
// Kernel 1: one workgroup (10 wave32s) per row. Single-pass online softmax:
// running (max m, sum s = sum exp(x-m)) + plain row sum. Each thread owns
// exactly 25 float4s; unrolled x5 so each trip issues a 5-wide clause of
// global_load_b128 (2.5KB in flight/wave) and does one rescale check per 20
// elements. Reads pred exactly once: 1.05GB -> ~45us floor @ 23.3 TB/s.
// ---------------------------------------------------------------------------
__global__ __launch_bounds__(BLOCK1)
void ls_row_loss_kernel(const float* __restrict__ pred,
                        const int*   __restrict__ target,
                        float*       __restrict__ row_loss)
{
    const int row  = blockIdx.x;
    const int tid  = threadIdx.x;
    const int lane = tid & 31;
    const int wave = tid >> 5;

    const size_t base = (size_t)row * (size_t)VOCAB;
    const float4* __restrict__ p4 = (const float4*)(pred + base);

    float m = -INFINITY;   // running max
    float s = 0.0f;        // running sum of exp(x - m)
    float r = 0.0f;        // plain row sum

    #pragma unroll
    for (int k = 0; k < 25; k += 5) {
        // 5-load clause; wave touches 32 consecutive float4 = 512B per load.
        float4 v0 = p4[tid + (k + 0) * BLOCK1];
        float4 v1 = p4[tid + (k + 1) * BLOCK1];
        float4 v2 = p4[tid + (k + 2) * BLOCK1];
        float4 v3 = p4[tid + (k + 3) * BLOCK1];
        float4 v4 = p4[tid + (k + 4) * BLOCK1];

        r += ((v0.x + v0.y) + (v0.z + v0.w))
           + ((v1.x + v1.y) + (v1.z + v1.w))
           + ((v2.x + v2.y) + (v2.z + v2.w))
           + ((v3.x + v3.y) + (v3.z + v3.w))
           + ((v4.x + v4.y) + (v4.z + v4.w));

        float vm = fmaxf(fmaxf(fmaxf(v0.x, v0.y), fmaxf(v0.z, v0.w)),
                   fmaxf(fmaxf(fmaxf(v1.x, v1.y), fmaxf(v1.z, v1.w)),
                   fmaxf(fmaxf(fmaxf(v2.x, v2.y), fmaxf(v2.z, v2.w)),
                   fmaxf(fmaxf(fmaxf(v3.x, v3.y), fmaxf(v3.z, v3.w)),
                         fmaxf(fmaxf(v4.x, v4.y), fmaxf(v4.z, v4.w))))));
        if (vm > m) {                 // at most one rescale per 20 elements
            s *= __expf(m - vm);      // m==-inf on first trip -> s stays 0
            m = vm;
        }
        s += (__expf(v0.x - m) + __expf(v0.y - m) + __expf(v0.z - m) + __expf(v0.w - m))
           + (__expf(v1.x - m) + __expf(v1.y - m) + __expf(v1.z - m) + __expf(v1.w - m))
           + (__expf(v2.x - m) + __expf(v2.y - m) + __expf(v2.z - m) + __expf(v2.w - m))
           + (__expf(v3.x - m) + __expf(v3.y - m) + __expf(v3.z - m) + __expf(v3.w - m))
           + (__expf(v4.x - m) + __expf(v4.y - m) + __expf(v4.z - m) + __expf(v4.w - m));
    }

    // ---- wave32 reduction of (m, s, r) via shuffles ----
    #pragma unroll
    for (int off = 16; off > 0; off >>= 1) {
        float mo = __shfl_xor(m, off, 32);
        float so = __shfl_xor(s, off, 32);
        float ro = __shfl_xor(r, off, 32);
        float M  = fmaxf(m, mo);
        s = s * __expf(m - M) + so * __expf(mo - M);
        m = M;
        r += ro;
    }

    // ---- cross-wave combine through LDS (10 waves) ----
    __shared__ float sm[NWAVES1], ss[NWAVES1], sr[NWAVES1];
    if (lane == 0) { sm[wave] = m; ss[wave] = s; sr[wave] = r; }
    __syncthreads();

    if (tid == 0) {
        float M = sm[0], S = ss[0], R = sr[0];
        #pragma unroll
        for (int w = 1; w < NWAVES1; ++w) {
            float mo = sm[w], so = ss[w];
            float Mn = fmaxf(M, mo);
            S = S * __expf(M - Mn) + so * __expf(mo - Mn);
            M = Mn;
            R += sr[w];
        }

        const int tgt = target[row];
        float loss = 0.0f;
        if (tgt != 0) {   // padding rows contribute zero
            float lse = M + __logf(S);
            float p0  = pred[base];                 // padding column logit
            float pt  = pred[base + (size_t)tgt];   // label logit
            // sum_j t_j log t_j  (constant for non-pad rows); (V-2)*eps == SMOOTHING
            float c1 = CONFIDENCE * __logf(CONFIDENCE)
                     + SMOOTHING * __logf(EPS_SMOOTH);
            // sum_j t_j * pred_j:
            float tdotp = EPS_SMOOTH * (R - p0 - pt) + CONFIDENCE * pt;
            // sum_j t_j == 1 for non-pad rows  ->  + lse * 1
            loss = c1 - tdotp + lse;
        }
        row_loss[row] = loss;
    }
}

// ---------------------------------------------------------------------------
// Kernel 2: reduce 8192 per-row losses to the scalar with
// V_WMMA_F32_16X16X4_F32 against an all-ones B matrix:
//   D[m][n] = sum_k A[m][k] + C[m][n]  -> 64 fp32 values summed per wmma,
// accumulated in fp32 across 16 iterations per wave (8 waves x 1024 values).
// Since we sum everything, the physical A layout is irrelevant: any bijection
// of 64 inputs into A gives the same total.
// ---------------------------------------------------------------------------
__global__ __launch_bounds__(BLOCK2)
void ls_final_reduce_kernel(const float* __restrict__ part,
                            float*       __restrict__ out)
{
    const int tid  = threadIdx.x;
    const int lane = tid & 31;
    const int wave = tid >> 5;

    const float* __restrict__ src = part + wave * 1024;

    v8f c = {0.f, 0.f, 0.f, 0.f, 0.f, 0.f, 0.f, 0.f};
    v2f b = {1.0f, 1.0f};

    #pragma unroll
    for (int it = 0; it < 16; ++it) {
        v2f a;
        a.x = src[it * 64 + lane * 2 + 0];
        a.y = src[it * 64 + lane * 2 + 1];
        // (neg_a, A, neg_b, B, c_mod, C, reuse_a, reuse_b)
        c = __builtin_amdgcn_wmma_f32_16x16x4_f32(
                false, a, false, b, (short)0, c, false, false);
    }

    // Each lane: sum its 8 D-matrix registers (rows m=j or m=j+8, col n=lane%16)
    float t = 0.0f;
    #pragma unroll
    for (int j = 0; j < 8; ++j) t += c[j];
    // Column n's full sum lives split across lane n and lane n+16:
    t += __shfl_xor(t, 16, 32);      // every lane now holds this wave's total

    __shared__ float swsum[8];
    if (lane == 0) swsum[wave] = t;
    __syncthreads();

    if (tid == 0) {
        float total = 0.0f;
        #pragma unroll
        for (int w = 0; w < 8; ++w) total += swsum[w];
        out[0] = total * (1.0f / (float)N_TOKENS);   // batchmean
    }
}

// ---------------------------------------------------------------------------
extern "C" void kernel_launch(void* const* d_in, const int* in_sizes, int n_in,
                              void* d_out, int out_size, void* d_ws, size_t ws_size,
                              hipStream_t stream)
{
    const float* pred   = (const float*)d_in[0];   // (8192, 32000) fp32
    const int*   target = (const int*)  d_in[1];   // (8192,)
    float* out      = (float*)d_out;               // scalar
    float* row_loss = (float*)d_ws;                // 8192 floats of scratch

    ls_row_loss_kernel<<<N_TOKENS, BLOCK1, 0, stream>>>(pred, target, row_loss);
    ls_final_reduce_kernel<<<1, BLOCK2, 0, stream>>>(row_loss, out);
}